// IntersectConcatAttention_52828097741217
// MI455X (gfx1250) — compile-verified
//
#include <hip/hip_runtime.h>

// IntersectConcatAttention fused kernel for gfx1250 (MI455X).
// One pass over key_embeds (256 MB) via register tiling; attention softmax
// reductions via LDS ds_add_f32; post projection done with
// V_WMMA_F32_16X16X4_F32 (exact f32 matrix path).

#define DIM    128
#define NPATH  8
#define NK     4
#define BATCH  65536
#define BT     32      // batch columns per workgroup
#define CH     16      // d-rows per thread (8 waves * 16 = 128)
#define WSTR   132     // padded LDS row stride (bank-conflict-free WMMA frags)

typedef float v2f __attribute__((ext_vector_type(2)));
typedef float v8f __attribute__((ext_vector_type(8)));

__launch_bounds__(256)
__global__ void ica_fused_kernel(const float* __restrict__ q,      // [128, B]
                                 const float* __restrict__ keys,   // [8, 128, B]
                                 const float* __restrict__ atten,  // [256, 4]
                                 const float* __restrict__ ln_g,   // [128]
                                 const float* __restrict__ ln_b,   // [128]
                                 const float* __restrict__ postW,  // [128, 128]
                                 const float* __restrict__ postB,  // [128]
                                 const float* __restrict__ pln_g,  // [128]
                                 const float* __restrict__ pln_b,  // [128]
                                 float* __restrict__ out)          // [128, B]
{
    __shared__ __align__(16) float lds_W[DIM * WSTR];     // 67584 B
    __shared__ __align__(16) float cT[BT * WSTR];         // 16896 B  (cT[col][d])
    __shared__ float lds_logit[NPATH * NK * BT];          // 4096 B
    __shared__ float lds_red[4 * BT];                     // 512 B

    const int t    = threadIdx.x;
    const int lane = t & 31;          // batch column within tile
    const int wid  = t >> 5;          // wave id == d-chunk id == GEMM M-tile
    const int d0   = wid * CH;
    const int bbase = blockIdx.x * BT;
    const long b    = (long)bbase + lane;

    // ---- stage post_W into padded LDS (coalesced float4) ----
    {
        const float4* src = (const float4*)postW;
#pragma unroll
        for (int i = 0; i < 16; ++i) {
            int e4 = i * 256 + t;                 // 0..4095 float4s
            float4 v = src[e4];
            int row = e4 >> 5;                    // 128 floats per row = 32 float4
            int col = (e4 & 31) << 2;
            float* dst = &lds_W[row * WSTR + col];
            dst[0] = v.x; dst[1] = v.y; dst[2] = v.z; dst[3] = v.w;
        }
    }
    // ---- zero LDS accumulators ----
#pragma unroll
    for (int i = 0; i < 4; ++i) lds_logit[i * 256 + t] = 0.0f;
    if (t < 4 * BT) lds_red[t] = 0.0f;
    __syncthreads();

    // =======================================================================
    // Phase 1: attention logits (keys stay in registers, read once from HBM)
    // =======================================================================
    float qreg[CH];
    float kreg[NPATH][CH];
    float qp[NK] = {0.f, 0.f, 0.f, 0.f};
    float lp[NPATH][NK];
#pragma unroll
    for (int n = 0; n < NPATH; ++n)
#pragma unroll
        for (int k = 0; k < NK; ++k) lp[n][k] = 0.0f;

#pragma unroll
    for (int dd = 0; dd < CH; ++dd) {
        const int d = d0 + dd;
        const float qv = q[(long)d * BATCH + b];
        qreg[dd] = qv;
#pragma unroll
        for (int k = 0; k < NK; ++k)
            qp[k] = fmaf(qv, atten[d * NK + k], qp[k]);           // top half (query)
        float ab[NK];
#pragma unroll
        for (int k = 0; k < NK; ++k) ab[k] = atten[(DIM + d) * NK + k]; // bottom half
#pragma unroll
        for (int n = 0; n < NPATH; ++n) {
            const float kv = keys[((long)n * DIM + d) * BATCH + b];
            kreg[n][dd] = kv;
#pragma unroll
            for (int k = 0; k < NK; ++k)
                lp[n][k] = fmaf(kv, ab[k], lp[n][k]);
        }
    }
    // cross-chunk reduction: sum over the 8 d-chunks. Each chunk also adds its
    // partial q·A_top, so the total per (n,k) is qa[k] + key_logit[n,k].
#pragma unroll
    for (int n = 0; n < NPATH; ++n)
#pragma unroll
        for (int k = 0; k < NK; ++k)
            atomicAdd(&lds_logit[(n * NK + k) * BT + lane], lp[n][k] + qp[k]);
    __syncthreads();

    // ---- leaky-relu + softmax over paths (redundant per chunk-thread) ----
    float w[NPATH] = {0.f, 0.f, 0.f, 0.f, 0.f, 0.f, 0.f, 0.f};
#pragma unroll
    for (int k = 0; k < NK; ++k) {
        float l[NPATH];
#pragma unroll
        for (int n = 0; n < NPATH; ++n) {
            float x = lds_logit[(n * NK + k) * BT + lane];
            l[n] = (x < 0.0f) ? 0.01f * x : x;                    // leaky_relu(0.01)
        }
        float m = l[0];
#pragma unroll
        for (int n = 1; n < NPATH; ++n) m = fmaxf(m, l[n]);
        float e[NPATH];
        float s = 0.0f;
#pragma unroll
        for (int n = 0; n < NPATH; ++n) { e[n] = __expf(l[n] - m); s += e[n]; }
        const float inv = 1.0f / s;
#pragma unroll
        for (int n = 0; n < NPATH; ++n) w[n] = fmaf(e[n], inv, w[n]);
    }
#pragma unroll
    for (int n = 0; n < NPATH; ++n) w[n] *= 0.25f;                // mean over k

    // ---- weighted combine (from registers) + relu + residual + LN1 ----
    float cpre[CH];
    {
        float s1 = 0.0f, s2 = 0.0f;
#pragma unroll
        for (int dd = 0; dd < CH; ++dd) {
            float a = 0.0f;
#pragma unroll
            for (int n = 0; n < NPATH; ++n) a = fmaf(w[n], kreg[n][dd], a);
            a = fmaxf(a, 0.0f) + qreg[dd];
            cpre[dd] = a;
            s1 += a;
            s2 = fmaf(a, a, s2);
        }
        atomicAdd(&lds_red[lane], s1);
        atomicAdd(&lds_red[BT + lane], s2);
    }
    __syncthreads();
    {
        const float mu  = lds_red[lane] * (1.0f / DIM);
        const float var = lds_red[BT + lane] * (1.0f / DIM) - mu * mu;
        const float rs  = rsqrtf(var + 1e-6f);
#pragma unroll
        for (int dd = 0; dd < CH; ++dd) {
            const int d = d0 + dd;
            cT[lane * WSTR + d] = ln_g[d] * (cpre[dd] - mu) * rs + ln_b[d];
        }
    }
    __syncthreads();

    // =======================================================================
    // Phase 2: out = LN2( post_W @ c + post_B + c )  via f32 WMMA
    // Wave `wid` -> M-tile rows [wid*16, wid*16+16); N-tiles: cols 0-15, 16-31
    // A frag (16x4 f32): lanes 0-15 hold K=0,1; lanes 16-31 hold K=2,3
    // B frag (4x16 f32): lanes 0-15 hold rows K=0,1; lanes 16-31 rows K=2,3
    // =======================================================================
    const int half = lane >> 4;
    const int hl   = lane & 15;
    v8f acc0 = {0.f, 0.f, 0.f, 0.f, 0.f, 0.f, 0.f, 0.f};
    v8f acc1 = {0.f, 0.f, 0.f, 0.f, 0.f, 0.f, 0.f, 0.f};
    const float* Wr  = &lds_W[(wid * 16 + hl) * WSTR + 2 * half];
    const float* c0p = &cT[hl * WSTR + 2 * half];
    const float* c1p = &cT[(hl + 16) * WSTR + 2 * half];
#pragma unroll
    for (int kk = 0; kk < 32; ++kk) {
        v2f a  = *(const v2f*)(Wr  + 4 * kk);
        v2f b0 = *(const v2f*)(c0p + 4 * kk);
        v2f b1 = *(const v2f*)(c1p + 4 * kk);
        acc0 = __builtin_amdgcn_wmma_f32_16x16x4_f32(false, a, false, b0,
                                                     (short)0, acc0, false, false);
        acc1 = __builtin_amdgcn_wmma_f32_16x16x4_f32(false, a, false, b1,
                                                     (short)0, acc1, false, false);
    }

    // ---- epilogue: +bias, +residual, LN2 reduction ----
    float v0[8], v1[8];
    {
        float s1a = 0.f, s2a = 0.f, s1b = 0.f, s2b = 0.f;
#pragma unroll
        for (int j = 0; j < 8; ++j) {
            const int row = wid * 16 + 8 * half + j;   // D-matrix row for VGPR j
            const float pb = postB[row];
            const float x0 = acc0[j] + pb + cT[hl * WSTR + row];
            const float x1 = acc1[j] + pb + cT[(hl + 16) * WSTR + row];
            v0[j] = x0; v1[j] = x1;
            s1a += x0; s2a = fmaf(x0, x0, s2a);
            s1b += x1; s2b = fmaf(x1, x1, s2b);
        }
        atomicAdd(&lds_red[2 * BT + hl],      s1a);
        atomicAdd(&lds_red[2 * BT + 16 + hl], s1b);
        atomicAdd(&lds_red[3 * BT + hl],      s2a);
        atomicAdd(&lds_red[3 * BT + 16 + hl], s2b);
    }
    __syncthreads();
    {
        const float mu0  = lds_red[2 * BT + hl] * (1.0f / DIM);
        const float var0 = lds_red[3 * BT + hl] * (1.0f / DIM) - mu0 * mu0;
        const float rs0  = rsqrtf(var0 + 1e-6f);
        const float mu1  = lds_red[2 * BT + 16 + hl] * (1.0f / DIM);
        const float var1 = lds_red[3 * BT + 16 + hl] * (1.0f / DIM) - mu1 * mu1;
        const float rs1  = rsqrtf(var1 + 1e-6f);
#pragma unroll
        for (int j = 0; j < 8; ++j) {
            const int row = wid * 16 + 8 * half + j;
            const float g  = pln_g[row];
            const float bb = pln_b[row];
            out[(long)row * BATCH + bbase + hl]      = g * (v0[j] - mu0) * rs0 + bb;
            out[(long)row * BATCH + bbase + 16 + hl] = g * (v1[j] - mu1) * rs1 + bb;
        }
    }
}

extern "C" void kernel_launch(void* const* d_in, const int* in_sizes, int n_in,
                              void* d_out, int out_size, void* d_ws, size_t ws_size,
                              hipStream_t stream) {
    (void)in_sizes; (void)n_in; (void)out_size; (void)d_ws; (void)ws_size;
    const float* q      = (const float*)d_in[0];
    const float* keys   = (const float*)d_in[1];
    const float* atten  = (const float*)d_in[2];
    const float* ln_g   = (const float*)d_in[3];
    const float* ln_b   = (const float*)d_in[4];
    const float* postW  = (const float*)d_in[5];
    const float* postB  = (const float*)d_in[6];
    const float* pln_g  = (const float*)d_in[7];
    const float* pln_b  = (const float*)d_in[8];
    float* out = (float*)d_out;

    dim3 grid(BATCH / BT);   // 2048 workgroups of 256 threads (8 wave32)
    dim3 block(256);
    ica_fused_kernel<<<grid, block, 0, stream>>>(q, keys, atten, ln_g, ln_b,
                                                 postW, postB, pln_g, pln_b, out);
}